// GCN_20151986553189
// MI455X (gfx1250) — compile-verified
//
#include <hip/hip_runtime.h>
#include <math.h>

typedef float v2f __attribute__((ext_vector_type(2)));
typedef float v4f __attribute__((ext_vector_type(4)));
typedef float v8f __attribute__((ext_vector_type(8)));

// ---------------------------------------------------------------------------
// Degree / normalization
// ---------------------------------------------------------------------------
__global__ void gcn_init_deg(float* __restrict__ deg, int n) {
    int i = blockIdx.x * blockDim.x + threadIdx.x;
    if (i < n) deg[i] = 1.0f;                       // self-loop contributes 1
}

__global__ void gcn_count_deg(const int* __restrict__ dst, float* __restrict__ deg,
                              long long n_edges) {
    long long e = (long long)blockIdx.x * blockDim.x + threadIdx.x;
    if (e < n_edges) {
        int d = __builtin_nontemporal_load(dst + e);   // streaming index read (NT)
        atomicAdd(&deg[d], 1.0f);
    }
}

__global__ void gcn_make_dinv(float* __restrict__ deg, int n) {
    int i = blockIdx.x * blockDim.x + threadIdx.x;
    if (i < n) deg[i] = rsqrtf(deg[i]);             // deg >= 1 always (self-loops)
}

// ---------------------------------------------------------------------------
// Layer 1 GEMM: xw = x @ W1   ([N,3] @ [3,16]) — one V_WMMA_F32_16X16X4_F32
// per 16 nodes, K=3 zero-padded to 4 (padding via cndmask selects, no branches).
// ---------------------------------------------------------------------------
__global__ void gcn_gemm1_wmma(const float* __restrict__ x, const float* __restrict__ W1,
                               float* __restrict__ xw, int n_nodes) {
    int wave = (int)((blockIdx.x * blockDim.x + threadIdx.x) >> 5);
    int lane = threadIdx.x & 31;
    int base = wave * 16;
    if (base >= n_nodes) return;                    // wave-uniform: EXEC stays all-ones
    int m  = lane & 15;
    int hi = lane >> 4;                             // 0: K=0,1  1: K=2,3
    int kb = hi * 2;
    int row  = base + m;
    int rowc = row < n_nodes ? row : n_nodes - 1;   // clamp (tail stores masked)
    int ncol = m;

    // A: lane holds (K=kb, K=kb+1) of row m; K=3 is the zero pad.
    float ax  = x[(long long)rowc * 3 + kb];
    float ayr = x[(long long)rowc * 3 + (hi ? 2 : 1)];   // clamped index, then select
    v2f a; a.x = ax; a.y = hi ? 0.0f : ayr;
    // B: W1[k][n], K=3 zero pad via select.
    float bx  = W1[kb * 16 + ncol];
    float byr = W1[(hi ? 2 : 1) * 16 + ncol];
    v2f b; b.x = bx; b.y = hi ? 0.0f : byr;

    v8f c = {};
    c = __builtin_amdgcn_wmma_f32_16x16x4_f32(false, a, false, b, (short)0, c, false, false);

    // C/D layout: VGPR r -> (M = r + 8*hi, N = lane&15)
    if (base + 16 <= n_nodes) {                     // full wave: clause of 8 stores
        float* p = xw + (long long)(base + hi * 8) * 16 + ncol;
#pragma unroll
        for (int r = 0; r < 8; ++r) p[r * 16] = c[r];
    } else {                                        // tail wave (generic path)
        for (int r = 0; r < 8; ++r) {
            int orow = base + r + hi * 8;
            if (orow < n_nodes) xw[(long long)orow * 16 + ncol] = c[r];
        }
    }
}

// ---------------------------------------------------------------------------
// Self-loop init (doubles as accumulator zero-init): agg = dinv^2 * feat
// Vectorized b128; W is compile-time so index math is shifts.
// ---------------------------------------------------------------------------
template <int W>
__global__ void gcn_selfloop_init(const float* __restrict__ dinv,
                                  const float* __restrict__ feat,
                                  float* __restrict__ agg, long long n_nodes) {
    constexpr int VPR = W / 4;                      // float4 chunks per row
    long long t = (long long)blockIdx.x * blockDim.x + threadIdx.x;
    if (t >= n_nodes * VPR) return;
    int i = (int)(t / VPR);                         // power-of-two shift
    float w = dinv[i] * dinv[i];
    v4f f = *(const v4f*)(feat + t * 4);
    v4f o; o.x = w * f.x; o.y = w * f.y; o.z = w * f.z; o.w = w * f.w;
    *(v4f*)(agg + t * 4) = o;
}

// ---------------------------------------------------------------------------
// Edge scatter: W lanes per edge, ACT active columns (<= W).
// Index streams are NT (read-once, 128MB) so they don't evict the 16..32MB
// feature arrays whose random gathers/atomics stay resident in the 192MB L2.
// ---------------------------------------------------------------------------
template <int W, int ACT>
__global__ void gcn_scatter(const int* __restrict__ src, const int* __restrict__ dst,
                            const float* __restrict__ dinv, const float* __restrict__ feat,
                            float* __restrict__ agg, long long n_edges) {
    long long t = (long long)blockIdx.x * blockDim.x + threadIdx.x;
    long long e = t / W;
    int j = (int)(t % W);
    if (e >= n_edges) return;
    if (ACT < W && j >= ACT) return;                // skip structurally-zero pad column
    int s = __builtin_nontemporal_load(src + e);
    int d = __builtin_nontemporal_load(dst + e);
    float w = dinv[s] * dinv[d];
    atomicAdd(&agg[(long long)d * W + j], w * feat[(long long)s * W + j]);
}

// ---------------------------------------------------------------------------
// Layer 2 GEMM fused with bias+ReLU: hw = relu(agg1 + b1) @ W2
// ([N,16] @ [16,7] padded to 16 cols) — chain of 4 WMMA 16x16x4 f32 (K=16).
// N-padding via cndmask selects; only first 8 output columns stored.
// ---------------------------------------------------------------------------
__global__ void gcn_gemm2_wmma(const float* __restrict__ agg1, const float* __restrict__ b1,
                               const float* __restrict__ W2, float* __restrict__ hw,
                               int n_nodes) {
    int wave = (int)((blockIdx.x * blockDim.x + threadIdx.x) >> 5);
    int lane = threadIdx.x & 31;
    int base = wave * 16;
    if (base >= n_nodes) return;                    // wave-uniform exit
    int m  = lane & 15;
    int hi = lane >> 4;
    int kb = hi * 2;
    int ncol = m;
    int colc = ncol < 7 ? ncol : 6;                 // clamped index for pad lanes
    bool act = ncol < 7;
    int row  = base + m;
    int rowc = row < n_nodes ? row : n_nodes - 1;
    const float* arow = agg1 + (long long)rowc * 16;

    v8f c = {};
#pragma unroll
    for (int t = 0; t < 4; ++t) {
        int k0 = 4 * t + kb;                        // even -> 8B-aligned b64 load
        v2f av = *(const v2f*)(arow + k0);
        v2f a, b;
        a.x = fmaxf(av.x + b1[k0], 0.0f);
        a.y = fmaxf(av.y + b1[k0 + 1], 0.0f);
        float bx = W2[k0 * 7 + colc];
        float by = W2[(k0 + 1) * 7 + colc];
        b.x = act ? bx : 0.0f;
        b.y = act ? by : 0.0f;
        c = __builtin_amdgcn_wmma_f32_16x16x4_f32(false, a, false, b, (short)0, c, false, false);
    }

    if (ncol < 8) {
        if (base + 16 <= n_nodes) {                 // full wave: clause of 8 stores
            float* p = hw + (long long)(base + hi * 8) * 8 + ncol;
#pragma unroll
            for (int r = 0; r < 8; ++r) p[r * 8] = c[r];
        } else {
            for (int r = 0; r < 8; ++r) {
                int orow = base + r + hi * 8;
                if (orow < n_nodes) hw[(long long)orow * 8 + ncol] = c[r];
            }
        }
    }
}

// ---------------------------------------------------------------------------
// Final bias + log_softmax over 7 classes (b128 row loads, NT output stores)
// ---------------------------------------------------------------------------
__global__ void gcn_logsoftmax(const float* __restrict__ agg2, const float* __restrict__ b2,
                               float* __restrict__ out, int n_nodes) {
    int i = blockIdx.x * blockDim.x + threadIdx.x;
    if (i >= n_nodes) return;
    const v4f* rowp = (const v4f*)(agg2 + (long long)i * 8);  // 32B-aligned row
    v4f lo = rowp[0];
    v4f hi4 = rowp[1];
    float v[7] = {lo.x, lo.y, lo.z, lo.w, hi4.x, hi4.y, hi4.z};
    float mx = -INFINITY;
#pragma unroll
    for (int j = 0; j < 7; ++j) {
        v[j] += b2[j];
        mx = fmaxf(mx, v[j]);
    }
    float s = 0.0f;
#pragma unroll
    for (int j = 0; j < 7; ++j) s += expf(v[j] - mx);
    float lse = mx + logf(s);
#pragma unroll
    for (int j = 0; j < 7; ++j)
        __builtin_nontemporal_store(v[j] - lse, out + (long long)i * 7 + j);
}

// ---------------------------------------------------------------------------
// Launch
// ---------------------------------------------------------------------------
extern "C" void kernel_launch(void* const* d_in, const int* in_sizes, int n_in,
                              void* d_out, int out_size, void* d_ws, size_t ws_size,
                              hipStream_t stream) {
    const float* x  = (const float*)d_in[0];
    const int*   ei = (const int*)d_in[1];
    const float* W1 = (const float*)d_in[2];
    const float* b1 = (const float*)d_in[3];
    const float* W2 = (const float*)d_in[4];
    const float* b2 = (const float*)d_in[5];

    const int       n = in_sizes[0] / 3;              // 500000
    const long long E = (long long)in_sizes[1] / 2;   // 16000000
    const int* src = ei;
    const int* dst = ei + E;

    // Workspace layout (floats), 256B-aligned chunks
    auto align64 = [](long long v) { return (v + 63) & ~63LL; };
    float* ws   = (float*)d_ws;
    long long off = 0;
    float* dinv = ws + off; off += align64(n);                  // deg -> dinv in place
    float* xw   = ws + off; off += align64((long long)n * 16);
    float* agg1 = ws + off; off += align64((long long)n * 16);
    float* hw   = ws + off; off += align64((long long)n * 8);
    float* agg2 = ws + off; off += align64((long long)n * 8);

    const int BLK = 256;
    auto blocks = [&](long long work) { return (unsigned)((work + BLK - 1) / BLK); };

    // normalization
    gcn_init_deg<<<blocks(n), BLK, 0, stream>>>(dinv, n);
    gcn_count_deg<<<blocks(E), BLK, 0, stream>>>(dst, dinv, E);
    gcn_make_dinv<<<blocks(n), BLK, 0, stream>>>(dinv, n);

    // layer 1
    long long waves = ((long long)n + 15) / 16;
    gcn_gemm1_wmma<<<blocks(waves * 32), BLK, 0, stream>>>(x, W1, xw, n);
    gcn_selfloop_init<16><<<blocks((long long)n * 4), BLK, 0, stream>>>(dinv, xw, agg1, n);
    gcn_scatter<16, 16><<<blocks(E * 16), BLK, 0, stream>>>(src, dst, dinv, xw, agg1, E);

    // layer 2 (bias+ReLU fused into the WMMA GEMM)
    gcn_gemm2_wmma<<<blocks(waves * 32), BLK, 0, stream>>>(agg1, b1, W2, hw, n);
    gcn_selfloop_init<8><<<blocks((long long)n * 2), BLK, 0, stream>>>(dinv, hw, agg2, n);
    gcn_scatter<8, 7><<<blocks(E * 8), BLK, 0, stream>>>(src, dst, dinv, hw, agg2, E);

    // output
    gcn_logsoftmax<<<blocks(n), BLK, 0, stream>>>(agg2, b2, (float*)d_out, n);
}